// PPScatter_23227183137502
// MI455X (gfx1250) — compile-verified
//
#include <hip/hip_runtime.h>

// PointPillars scatter for MI455X (gfx1250, wave32).
// Pure data movement: 256MB NT zero-fill + 12MB coalesced gather / scattered store.
// Roofline: ~280MB @ 23.3 TB/s ~= 12us. No matrix math -> no WMMA applicable;
// CDNA5 paths used: b128 non-temporal global stores (8x unrolled with immediate
// offsets), global_prefetch_b8, wave32 block sizing.

#define B_  4
#define C_  64
#define P_  12000
#define H_  512
#define W_  512

#define FILL_THREADS 256   // 8 wave32 per block
#define FILL_UNROLL  8     // 8 x b128 per thread -> 32KB contiguous per block

typedef __attribute__((ext_vector_type(4))) float v4f;

// ---------------------------------------------------------------------------
// Kernel 1: zero the 256MB output with 128-bit non-temporal stores.
// Each block owns a contiguous 32KB chunk; each thread issues 8 b128 NT
// stores at compile-time immediate offsets (u * 256 lanes * 16B = u * 4KB),
// so the address is computed once and the loop body is 8 back-to-back
// global_store_b128 ... th:TH_STORE_NT instructions. Adjacent lanes write
// adjacent 16B -> every store is a fully coalesced 512B wave transaction.
// ---------------------------------------------------------------------------
__global__ void pp_zero_b128_nt(float* __restrict__ out, long n4) {
    const v4f z = {0.0f, 0.0f, 0.0f, 0.0f};
    v4f* __restrict__ o = (v4f*)out;

    long base = (long)blockIdx.x * (FILL_THREADS * FILL_UNROLL) + threadIdx.x;

    if (base + (long)(FILL_UNROLL - 1) * FILL_THREADS < n4) {
        // Fast path (always taken with exact-fit grid): unguarded 8x unroll.
#pragma unroll
        for (int u = 0; u < FILL_UNROLL; ++u) {
            __builtin_nontemporal_store(z, o + base + u * FILL_THREADS);
        }
    } else {
        // Guarded tail for robustness to size changes.
#pragma unroll
        for (int u = 0; u < FILL_UNROLL; ++u) {
            long idx = base + (long)u * FILL_THREADS;
            if (idx < n4) __builtin_nontemporal_store(z, o + idx);
        }
    }
}

// ---------------------------------------------------------------------------
// Kernel 2: scatter x[b, c, p] -> out[b, c, yi, xi] for valid pillars.
// Flat work id i maps to (c, b, p) with p fastest, so adjacent lanes read
// adjacent p: x and inds loads are fully coalesced. Writes are inherently
// scattered 4B stores (only 12MB total). Grid-stride loop lets us prefetch
// the next iteration's index triple via global_prefetch_b8.
// ---------------------------------------------------------------------------
__global__ void pp_scatter(const float* __restrict__ x,
                           const int*   __restrict__ inds,
                           float*       __restrict__ out,
                           int total) {
    const int NP = B_ * P_;  // 48000 pillars
    int i      = blockIdx.x * blockDim.x + threadIdx.x;
    int stride = gridDim.x * blockDim.x;

    for (; i < total; i += stride) {
        // Prefetch next iteration's index triple (gfx1250 global_prefetch_b8).
        int nxt = i + stride;
        if (nxt < total) {
            int bp_n = nxt % NP;
            __builtin_prefetch(&inds[bp_n * 3], 0, 1);
        }

        int c  = i / NP;        // channel 0..63
        int bp = i - c * NP;    // flat pillar id = b*P + p
        int b  = bp / P_;
        int p  = bp - b * P_;

        int i0 = inds[bp * 3 + 0];
        if (i0 != 0) {
            int xi = inds[bp * 3 + 1];
            int yi = inds[bp * 3 + 2];
            if ((unsigned)xi < (unsigned)W_ && (unsigned)yi < (unsigned)H_) {
                float v = x[(b * C_ + c) * P_ + p];            // coalesced read
                out[((b * C_ + c) * H_ + yi) * W_ + xi] = v;   // scattered 4B store
            }
        }
    }
}

extern "C" void kernel_launch(void* const* d_in, const int* in_sizes, int n_in,
                              void* d_out, int out_size, void* d_ws, size_t ws_size,
                              hipStream_t stream) {
    const float* x    = (const float*)d_in[0];   // (B, C, P) f32
    const int*   inds = (const int*)d_in[1];     // (B, P, 3) i32
    float*       out  = (float*)d_out;           // (B, C, H, W) f32

    // --- Pass 1: zero-fill (stream-ordered before scatter) ---
    long n4 = (long)out_size / 4;                            // 16,777,216 float4s
    long per_block = FILL_THREADS * FILL_UNROLL;             // 2048 float4s = 32KB
    int  fill_blocks = (int)((n4 + per_block - 1) / per_block);  // 8192 (exact fit)
    pp_zero_b128_nt<<<fill_blocks, FILL_THREADS, 0, stream>>>(out, n4);

    // --- Pass 2: scatter ---
    int total = B_ * P_ * C_;                    // 3,072,000 (pillar, channel) pairs
    // 6000 blocks * 256 threads -> 2 grid-stride iterations per thread
    pp_scatter<<<6000, 256, 0, stream>>>(x, inds, out, total);
}